// GCN_81621558493696
// MI455X (gfx1250) — compile-verified
//
#include <hip/hip_runtime.h>
#include <hip/hip_bf16.h>

typedef __attribute__((ext_vector_type(2))) float v2f;
typedef __attribute__((ext_vector_type(8))) float v8f;

#define HID 64

// ---------------- utility kernels ----------------

__global__ void fill_f(float* __restrict__ p, float v, long long n) {
    long long i = (long long)blockIdx.x * blockDim.x + threadIdx.x;
    if (i < n) p[i] = v;
}

__global__ void deg_edges(const int* __restrict__ eidx, float* __restrict__ deg, int E) {
    long long e = (long long)blockIdx.x * blockDim.x + threadIdx.x;
    if (e < E) atomicAdd(&deg[eidx[(long long)E + e]], 1.0f);
}

__global__ void rsqrt_ip(float* __restrict__ p, int n) {
    int i = blockIdx.x * blockDim.x + threadIdx.x;
    if (i < n) p[i] = rsqrtf(p[i]);
}

// ---------------- dense transform: tmp = (relu?)h @ W  (N x 64 @ 64 x 64) ----------------
// One wave computes a 16-row x 64-col output tile with V_WMMA_F32_16X16X4_F32.
// W is staged in LDS pre-swizzled into B-fragment layout so each fragment is one
// ds_load_b64. RELU is a template parameter (no per-element cndmask).
// 8 waves / block => 128 rows per block. Requires nrows % 16 == 0 (100000 = 6250*16).

template <int RELU>
__global__ __launch_bounds__(256) void gcn_xform_wmma(
    const float* __restrict__ hin, const float* __restrict__ W,
    float* __restrict__ hout, int nrows)
{
    // B-fragment table: [kstep(16)][ntile(4)][lane(32)] -> (b.x, b.y)
    __shared__ v2f Wf[16 * 4 * 32];          // 16 KB of the 320 KB WGP LDS

    int tid = threadIdx.x;
    // stage + swizzle W: 2048 fragments, 8 per thread
    for (int idx = tid; idx < 16 * 4 * 32; idx += 256) {
        int lane_s = idx & 31;
        int nt     = (idx >> 5) & 3;
        int ks     = idx >> 7;
        int koff_s = (lane_s < 16) ? 0 : 2;
        int nsel_s = lane_s & 15;
        int k = ks * 4 + koff_s;
        int n = nt * 16 + nsel_s;
        v2f f;
        f.x = W[(size_t)k * HID + n];
        f.y = W[(size_t)(k + 1) * HID + n];
        Wf[idx] = f;
    }
    __syncthreads();

    int wave = tid >> 5;
    int lane = tid & 31;
    int tile = blockIdx.x * 8 + wave;        // 16-row tile index
    int row0 = tile * 16;
    if (row0 >= nrows) return;

    int m    = lane & 15;                    // A-matrix M (both lane halves)
    int koff = (lane < 16) ? 0 : 2;          // lanes 0-15: K=k0+0/1, lanes 16-31: K=k0+2/3
    int nsel = lane & 15;                    // B/C/D column within 16-wide tile

    v8f acc0 = {}, acc1 = {}, acc2 = {}, acc3 = {};
    const float* __restrict__ arow = hin + (size_t)(row0 + m) * HID + koff;
    const v2f* __restrict__ fb = Wf + lane;

    #pragma unroll
    for (int ks = 0; ks < 16; ks++) {
        // A fragment: one 8-byte load (k0+koff is even -> aligned)
        v2f a = *(const v2f*)(arow + ks * 4);
        if (RELU) { a.x = fmaxf(a.x, 0.0f); a.y = fmaxf(a.y, 0.0f); }

        v2f b0v = fb[(ks * 4 + 0) * 32];
        v2f b1v = fb[(ks * 4 + 1) * 32];
        v2f b2v = fb[(ks * 4 + 2) * 32];
        v2f b3v = fb[(ks * 4 + 3) * 32];

        acc0 = __builtin_amdgcn_wmma_f32_16x16x4_f32(false, a, false, b0v, (short)0, acc0, false, false);
        acc1 = __builtin_amdgcn_wmma_f32_16x16x4_f32(false, a, false, b1v, (short)0, acc1, false, false);
        acc2 = __builtin_amdgcn_wmma_f32_16x16x4_f32(false, a, false, b2v, (short)0, acc2, false, false);
        acc3 = __builtin_amdgcn_wmma_f32_16x16x4_f32(false, a, false, b3v, (short)0, acc3, false, false);
    }

    // C/D layout: VGPR r holds row r (lanes 0-15) / row r+8 (lanes 16-31), col = nsel
    int rbase = row0 + ((lane < 16) ? 0 : 8);
    #pragma unroll
    for (int r = 0; r < 8; r++) {
        float* __restrict__ orow = hout + (size_t)(rbase + r) * HID;
        orow[ 0 + nsel] = acc0[r];
        orow[16 + nsel] = acc1[r];
        orow[32 + nsel] = acc2[r];
        orow[48 + nsel] = acc3[r];
    }
}

// ---------------- epilogue: out[i,f] = tmp[i,f]*dinv[i]^2 + b[f] (self-loop + bias) -----

__global__ void gcn_init_out(const float* __restrict__ tmp, const float* __restrict__ dinv,
                             const float* __restrict__ bias, float* __restrict__ out,
                             long long n) {
    long long i = (long long)blockIdx.x * blockDim.x + threadIdx.x;
    if (i >= n) return;
    int node = (int)(i >> 6);
    int f = (int)(i & 63);
    float d = dinv[node];
    out[i] = tmp[i] * d * d + bias[f];
}

// ---------------- edge scatter: out[col] += tmp[row] * dinv[row]*dinv[col] --------------
// 16 work-items per edge, each handles a float4 chunk of the 64-wide feature row.

__global__ __launch_bounds__(256) void gcn_scatter(
    const float* __restrict__ tmp, const int* __restrict__ eidx,
    const float* __restrict__ dinv, float* __restrict__ out, int E)
{
    long long item = (long long)blockIdx.x * 256 + threadIdx.x;
    long long e = item >> 4;
    if (e >= E) return;
    int c4 = (int)(item & 15) * 4;
    int r = eidx[e];
    int c = eidx[(long long)E + e];
    float nrm = dinv[r] * dinv[c];
    const float4 v = *(const float4*)(tmp + (size_t)r * HID + c4);
    float* o = out + (size_t)c * HID + c4;
    atomicAdd(o + 0, v.x * nrm);
    atomicAdd(o + 1, v.y * nrm);
    atomicAdd(o + 2, v.z * nrm);
    atomicAdd(o + 3, v.w * nrm);
}

// ---------------- pooling ----------------

__global__ void pool_acc(const float* __restrict__ h, const int* __restrict__ batch,
                         float* __restrict__ pooled, float* __restrict__ cnt, long long n) {
    long long i = (long long)blockIdx.x * blockDim.x + threadIdx.x;
    if (i >= n) return;
    int node = (int)(i >> 6);
    int f = (int)(i & 63);
    int g = batch[node];
    atomicAdd(&pooled[(size_t)g * HID + f], h[i]);
    if (f == 0) atomicAdd(&cnt[g], 1.0f);
}

__global__ void final_lin(const float* __restrict__ pooled, const float* __restrict__ cnt,
                          const float* __restrict__ lw, const float* __restrict__ lb,
                          float* __restrict__ out, int G) {
    int i = blockIdx.x * blockDim.x + threadIdx.x;
    if (i >= G * 2) return;
    int g = i >> 1, o = i & 1;
    float inv = 1.0f / fmaxf(cnt[g], 1.0f);
    float s = 0.0f;
    #pragma unroll
    for (int k = 0; k < HID; k++) s += pooled[(size_t)g * HID + k] * lw[k * 2 + o];
    out[i] = s * inv + lb[o];
}

// ---------------- launcher ----------------

extern "C" void kernel_launch(void* const* d_in, const int* in_sizes, int n_in,
                              void* d_out, int out_size, void* d_ws, size_t ws_size,
                              hipStream_t stream) {
    const float* x     = (const float*)d_in[0];
    const int*   eidx  = (const int*)  d_in[1];
    const int*   batch = (const int*)  d_in[2];
    const float* W0 = (const float*)d_in[3];
    const float* b0 = (const float*)d_in[4];
    const float* W1 = (const float*)d_in[5];
    const float* b1 = (const float*)d_in[6];
    const float* W2 = (const float*)d_in[7];
    const float* b2 = (const float*)d_in[8];
    const float* lW = (const float*)d_in[9];
    const float* lb = (const float*)d_in[10];
    float* outp = (float*)d_out;

    const int N = in_sizes[0] / HID;          // 100000
    const int E = in_sizes[1] / 2;            // 1600000
    const int G = out_size / 2;               // 256

    // workspace layout (floats)
    float* ws     = (float*)d_ws;
    float* dinv   = ws;
    float* bufA   = ws + (size_t)((N + 255) & ~255);     // tmp (transform output)
    float* bufB   = bufA + (size_t)N * HID;              // aggregated h
    float* pooled = bufB + (size_t)N * HID;              // [G,64] then [G] counts
    float* cnt    = pooled + (size_t)G * HID;

    const long long NF = (long long)N * HID;
    dim3 blk(256);
    int gN   = (N + 255) / 256;
    int gE   = (E + 255) / 256;
    int gNF  = (int)((NF + 255) / 256);
    long long sItems = (long long)E * 16;
    int gS   = (int)((sItems + 255) / 256);
    int tiles = (N + 15) / 16;
    int gX   = (tiles + 7) / 8;

    // degrees -> dinv (self-loop counted via init 1.0)
    fill_f<<<gN, blk, 0, stream>>>(dinv, 1.0f, N);
    deg_edges<<<gE, blk, 0, stream>>>(eidx, dinv, E);
    rsqrt_ip<<<gN, blk, 0, stream>>>(dinv, N);

    // layer 0: x -> bufA -> bufB
    gcn_xform_wmma<0><<<gX, blk, 0, stream>>>(x, W0, bufA, N);
    gcn_init_out  <<<gNF, blk, 0, stream>>>(bufA, dinv, b0, bufB, NF);
    gcn_scatter   <<<gS, blk, 0, stream>>>(bufA, eidx, dinv, bufB, E);

    // layer 1 (relu on input): bufB -> bufA -> bufB
    gcn_xform_wmma<1><<<gX, blk, 0, stream>>>(bufB, W1, bufA, N);
    gcn_init_out  <<<gNF, blk, 0, stream>>>(bufA, dinv, b1, bufB, NF);
    gcn_scatter   <<<gS, blk, 0, stream>>>(bufA, eidx, dinv, bufB, E);

    // layer 2 (relu on input, no relu on output)
    gcn_xform_wmma<1><<<gX, blk, 0, stream>>>(bufB, W2, bufA, N);
    gcn_init_out  <<<gNF, blk, 0, stream>>>(bufA, dinv, b2, bufB, NF);
    gcn_scatter   <<<gS, blk, 0, stream>>>(bufA, eidx, dinv, bufB, E);

    // global mean pool + final linear
    fill_f<<<(G * HID + G + 255) / 256, blk, 0, stream>>>(pooled, 0.0f, (long long)G * HID + G);
    pool_acc<<<gNF, blk, 0, stream>>>(bufB, batch, pooled, cnt, NF);
    final_lin<<<(G * 2 + 255) / 256, blk, 0, stream>>>(pooled, cnt, lW, lb, outp, G);
}